// ConsistencyLoss_66468913873687
// MI455X (gfx1250) — compile-verified
//
#include <hip/hip_runtime.h>

typedef __attribute__((ext_vector_type(2))) float v2f;
typedef __attribute__((ext_vector_type(8))) float v8f;

#define WDIM 128
#define HDIM 128
#define DDIM 128
#define BDIM 4
#define NCH  3

constexpr int   N_PER_CH = WDIM * HDIM * DDIM;      // 2^21 elems per (b,ch)
constexpr int   VOX      = BDIM * N_PER_CH;         // 2^23 voxels total
constexpr int   BLOCK    = 256;                     // 8 wave32s
constexpr int   UNROLL   = 4;
constexpr int   GRID1    = VOX / (BLOCK * UNROLL);  // 8192 blocks (exact tile)
constexpr float INV_CNT  = 1.0f / (float)((long long)BDIM * NCH * N_PER_CH);

__device__ __forceinline__ int clampi(int v, int lo, int hi) {
    return v < lo ? lo : (v > hi ? hi : v);
}

// Exact fp32 wave reduction via V_WMMA_F32_16X16X4_F32.
// A = [lane value, 0] striped as 16x4, B = ones(4x16)  =>  D[m,n] = sum_k A[m,k].
// Lane l (l<16) holds D[0..7, l] in c[0..7]; lane l+16 holds D[8..15, l].
// sum(c) + xor16 partner = full column sum = sum of all 32 lane values.
// Requires EXEC == all ones (guaranteed: no divergence before the call).
__device__ __forceinline__ float wave_sum_wmma(float x) {
    v2f a; a[0] = x;    a[1] = 0.0f;
    v2f b; b[0] = 1.0f; b[1] = 1.0f;
    v8f c = {};
    c = __builtin_amdgcn_wmma_f32_16x16x4_f32(false, a, false, b,
                                              (short)0, c, false, false);
    float s = c[0] + c[1] + c[2] + c[3] + c[4] + c[5] + c[6] + c[7];
    s += __shfl_xor(s, 16, 32);
    return s;
}

__global__ void __launch_bounds__(BLOCK)
warp_mse_partial(const float* __restrict__ s_ddf,
                 const float* __restrict__ t_ddf,
                 const float* __restrict__ a_ddf,
                 float* __restrict__ partial) {
    const int tid   = threadIdx.x;
    const int chunk = blockIdx.x * (BLOCK * UNROLL);  // 1024 consecutive voxels
    float acc = 0.0f;

    #pragma unroll
    for (int u = 0; u < UNROLL; ++u) {
        const int v = chunk + u * BLOCK + tid;   // global voxel id, coalesced in z
        const int b = v >> 21;                   // / N_PER_CH
        const int r = v & (N_PER_CH - 1);        // offset within (b,ch) volume
        const int x = r >> 14;                   // / (H*D)
        const int y = (r >> 7) & 127;
        const int z = r & 127;

        const size_t boff = (size_t)b * NCH * N_PER_CH;
        const float* __restrict__ ab = a_ddf + boff;
        const float* __restrict__ tb = t_ddf + boff;
        const float* __restrict__ sb = s_ddf + boff;

        if (u + 1 < UNROLL) {   // speculative prefetch of next streaming lines
            __builtin_prefetch(ab + r + BLOCK, 0, 0);
            __builtin_prefetch(tb + r + BLOCK, 0, 0);
        }

        // Streamed exactly once -> non-temporal so s_ddf stays resident in L2.
        const float a0 = __builtin_nontemporal_load(ab + r);
        const float a1 = __builtin_nontemporal_load(ab + N_PER_CH + r);
        const float a2 = __builtin_nontemporal_load(ab + 2 * N_PER_CH + r);

        const float cx = (float)x + a0;
        const float cy = (float)y + a1;
        const float cz = (float)z + a2;
        const float fx = floorf(cx), fy = floorf(cy), fz = floorf(cz);
        const float wx = cx - fx,    wy = cy - fy,    wz = cz - fz;

        const int ix0 = clampi((int)fx,     0, WDIM - 1);
        const int ix1 = clampi((int)fx + 1, 0, WDIM - 1);
        const int iy0 = clampi((int)fy,     0, HDIM - 1);
        const int iy1 = clampi((int)fy + 1, 0, HDIM - 1);
        const int iz0 = clampi((int)fz,     0, DDIM - 1);
        const int iz1 = clampi((int)fz + 1, 0, DDIM - 1);

        // Row offsets shared by all 3 channels.
        const int o00 = (ix0 * HDIM + iy0) * DDIM;
        const int o01 = (ix0 * HDIM + iy1) * DDIM;
        const int o10 = (ix1 * HDIM + iy0) * DDIM;
        const int o11 = (ix1 * HDIM + iy1) * DDIM;

        const float av[3] = {a0, a1, a2};
        #pragma unroll
        for (int ch = 0; ch < NCH; ++ch) {
            const float* __restrict__ sc = sb + ch * N_PER_CH;
            const float v000 = sc[o00 + iz0], v001 = sc[o00 + iz1];
            const float v010 = sc[o01 + iz0], v011 = sc[o01 + iz1];
            const float v100 = sc[o10 + iz0], v101 = sc[o10 + iz1];
            const float v110 = sc[o11 + iz0], v111 = sc[o11 + iz1];

            const float c00 = fmaf(wz, v001 - v000, v000);
            const float c01 = fmaf(wz, v011 - v010, v010);
            const float c10 = fmaf(wz, v101 - v100, v100);
            const float c11 = fmaf(wz, v111 - v110, v110);
            const float c0  = fmaf(wy, c01 - c00, c00);
            const float c1  = fmaf(wy, c11 - c10, c10);
            const float warped = fmaf(wx, c1 - c0, c0);

            const float tv   = __builtin_nontemporal_load(tb + ch * N_PER_CH + r);
            const float diff = av[ch] + warped - tv;
            acc = fmaf(diff, diff, acc);
        }
    }

    // Wave reduction on the matrix path (EXEC is all ones here).
    const float wsum = wave_sum_wmma(acc);

    __shared__ float lds[BLOCK / 32];
    const int lane = tid & 31, wave = tid >> 5;
    if (lane == 0) lds[wave] = wsum;
    __syncthreads();
    if (tid == 0) {
        float t = 0.0f;
        #pragma unroll
        for (int i = 0; i < BLOCK / 32; ++i) t += lds[i];
        partial[blockIdx.x] = t;
    }
}

__global__ void __launch_bounds__(BLOCK)
final_reduce(const float* __restrict__ partial, float* __restrict__ out) {
    const int tid = threadIdx.x;
    float acc = 0.0f;
    // GRID1 (8192) is a multiple of BLOCK -> no divergence, EXEC stays full.
    for (int i = tid; i < GRID1; i += BLOCK)
        acc += partial[i];

    const float wsum = wave_sum_wmma(acc);

    __shared__ float lds[BLOCK / 32];
    const int lane = tid & 31, wave = tid >> 5;
    if (lane == 0) lds[wave] = wsum;
    __syncthreads();
    if (tid == 0) {
        float t = 0.0f;
        #pragma unroll
        for (int i = 0; i < BLOCK / 32; ++i) t += lds[i];
        out[0] = t * INV_CNT;
    }
}

extern "C" void kernel_launch(void* const* d_in, const int* in_sizes, int n_in,
                              void* d_out, int out_size, void* d_ws, size_t ws_size,
                              hipStream_t stream) {
    const float* s_ddf = (const float*)d_in[0];   // (4,3,128,128,128) f32
    const float* t_ddf = (const float*)d_in[1];
    const float* a_ddf = (const float*)d_in[2];
    float* partial = (float*)d_ws;                // GRID1 floats = 32 KB scratch
    float* out     = (float*)d_out;               // scalar

    warp_mse_partial<<<GRID1, BLOCK, 0, stream>>>(s_ddf, t_ddf, a_ddf, partial);
    final_reduce<<<1, BLOCK, 0, stream>>>(partial, out);
}